// PolarFGDecoder_2388001817102
// MI455X (gfx1250) — compile-verified
//
#include <hip/hip_runtime.h>

// Polar-code min-sum BP decoder, fully LDS-resident per codeword.
// CODE_LEN=1024, INFO_LEN=512, N_STAGES=10, ITERS=5, CLIP=15.

#define N_CODE   1024
#define N_INFO   512
#define N_STG    10
#define N_ITERS  5
#define CLIPV    15.0f
#define TPB      256

#if defined(__has_builtin)
#  if __has_builtin(__builtin_amdgcn_global_load_async_to_lds_b128)
#    define HAVE_ASYNC_LD 1
#  endif
#  if __has_builtin(__builtin_amdgcn_global_store_async_from_lds_b32)
#    define HAVE_ASYNC_ST 1
#  endif
#  if __has_builtin(__builtin_amdgcn_fmed3f)
#    define HAVE_MED3 1
#  endif
#endif
#ifndef HAVE_ASYNC_LD
#  define HAVE_ASYNC_LD 0
#endif
#ifndef HAVE_ASYNC_ST
#  define HAVE_ASYNC_ST 0
#endif
#ifndef HAVE_MED3
#  define HAVE_MED3 0
#endif

typedef int v4i __attribute__((ext_vector_type(4)));
typedef __attribute__((address_space(1))) v4i* gv4i_p;   // global (AS1) int4*
typedef __attribute__((address_space(3))) v4i* lv4i_p;   // LDS (AS3) int4*
typedef __attribute__((address_space(1))) int* gi32_p;   // global (AS1) int*
typedef __attribute__((address_space(3))) int* li32_p;   // LDS (AS3) int*

__device__ __forceinline__ void wait_async0() {
#if defined(__has_builtin) && __has_builtin(__builtin_amdgcn_s_wait_asynccnt)
    __builtin_amdgcn_s_wait_asynccnt(0);
#else
    asm volatile("s_wait_asynccnt 0" ::: "memory");
#endif
}

// clamp to [-15, 15] in a single v_med3_f32 (inputs are finite here)
__device__ __forceinline__ float clip15(float x) {
#if HAVE_MED3
    return __builtin_amdgcn_fmed3f(x, -CLIPV, CLIPV);
#else
    return __builtin_fminf(__builtin_fmaxf(x, -CLIPV), CLIPV);
#endif
}

// f(a,b) = sign(a)*sign(b)*min(|a|,|b|); |.| folds into v_min_num source
// modifiers, and the (sgn & M) | (m & ~M) recombine pattern maps to v_bfi_b32.
__device__ __forceinline__ float minsum(float a, float b) {
    unsigned int sgn = __float_as_uint(a) ^ __float_as_uint(b);
    unsigned int m   = __float_as_uint(__builtin_fminf(__builtin_fabsf(a), __builtin_fabsf(b)));
    return __uint_as_float((sgn & 0x80000000u) | (m & 0x7fffffffu));
}

extern "C" __global__ __launch_bounds__(TPB)
void polar_bp_fg_kernel(const float* __restrict__ rx,
                        const int*   __restrict__ info_idx,
                        float*       __restrict__ out,
                        int batch)
{
    extern __shared__ float smem[];
    float* L    = smem;                          // [11][1024] left messages (stage 0 at L[0..1023])
    float* R    = smem + 11 * N_CODE;            // [11][1024] right messages
    int*   idxs = (int*)(smem + 22 * N_CODE);    // [512] info indices

    const int b   = blockIdx.x;
    const int tid = threadIdx.x;

    // ---------------- issue DMAs first (overlap with LDS fills) ----------------
#if HAVE_ASYNC_LD
    {
        // rx row -> L stage 10 (4 KB: 256 lanes x 16 B)
        v4i* gsrc = (v4i*)const_cast<float*>(rx + (size_t)b * N_CODE + tid * 4);
        v4i* ldst = (v4i*)&L[N_STG * N_CODE + tid * 4];
        __builtin_amdgcn_global_load_async_to_lds_b128((gv4i_p)gsrc, (lv4i_p)ldst, 0, 0);
        // info_idx -> idxs (2 KB: 128 lanes x 16 B)
        if (tid < N_INFO / 4) {
            v4i* gi = (v4i*)const_cast<int*>(info_idx + tid * 4);
            v4i* li = (v4i*)&idxs[tid * 4];
            __builtin_amdgcn_global_load_async_to_lds_b128((gv4i_p)gi, (lv4i_p)li, 0, 0);
        }
    }
#else
    *(float4*)&L[N_STG * N_CODE + tid * 4] = *(const float4*)(rx + (size_t)b * N_CODE + tid * 4);
    if (tid < N_INFO / 4)
        *(int4*)&idxs[tid * 4] = *(const int4*)(info_idx + tid * 4);
#endif

    // fills overlap the DMAs
#pragma unroll
    for (int i = tid; i < N_CODE; i += TPB) R[i] = CLIPV;          // right stage 0 (frozen = CLIP)
#pragma unroll
    for (int i = tid; i < (N_STG - 1) * N_CODE; i += TPB)          // left stages 1..9 = 0
        L[N_CODE + i] = 0.0f;

#if HAVE_ASYNC_LD
    wait_async0();
#endif
    __syncthreads();

    // scatter zeros at info positions of right stage 0
#pragma unroll
    for (int k = tid; k < N_INFO; k += TPB) R[idxs[k]] = 0.0f;

    // ---------------- BP iterations ----------------
    for (int it = 0; it < N_ITERS; ++it) {
        // right sweep: R[s+1] <- F(R[s], L[s+1]),  s = 0..9
#pragma unroll
        for (int s = 0; s < N_STG; ++s) {
            __syncthreads();
            const float* rs = R + s * N_CODE;
            float*       rd = R + (s + 1) * N_CODE;
            const float* ls = L + (s + 1) * N_CODE;
            const int st = 1 << s;
#pragma unroll
            for (int q = 0; q < 2; ++q) {
                const int p  = tid + q * TPB;                     // pair id 0..511
                const int iu = ((p >> s) << (s + 1)) | (p & (st - 1));
                const int il = iu + st;
                const float a  = rs[iu], c  = rs[il];
                const float lu = ls[iu], ll = ls[il];
                rd[iu] = clip15(minsum(a, ll + c));
                rd[il] = clip15(minsum(a, lu) + c);
            }
        }
        // left sweep: L[s] <- G(L[s+1], R[s]),  s = 9..0
#pragma unroll
        for (int s = N_STG - 1; s >= 0; --s) {
#if HAVE_ASYNC_ST
            // before anyone overwrites L stage 0, drain this wave's pending
            // async output stores (issued from L stage 0 last epilogue)
            if (s == 0) wait_async0();
#endif
            __syncthreads();
            const float* lsrc = L + (s + 1) * N_CODE;
            float*       ld   = L + s * N_CODE;
            const float* rs   = R + s * N_CODE;
            const int st = 1 << s;
#pragma unroll
            for (int q = 0; q < 2; ++q) {
                const int p  = tid + q * TPB;
                const int iu = ((p >> s) << (s + 1)) | (p & (st - 1));
                const int il = iu + st;
                const float lu = lsrc[iu], ll = lsrc[il];
                const float ru = rs[iu],   rl = rs[il];
                ld[iu] = clip15(minsum(lu, ll + rl));
                ld[il] = clip15(minsum(ru, lu) + ll);
            }
        }
        __syncthreads();

        // emit u[info_idx]: right[0] is 0 at every info position and never
        // rewritten, so u[idx] == L0[idx], which already lives in LDS.
        const size_t obase = ((size_t)it * batch + b) * N_INFO;
        const size_t odup  = (size_t)batch * N_INFO;   // offset of duplicated slot 5
#if HAVE_ASYNC_ST
#pragma unroll
        for (int k = tid; k < N_INFO; k += TPB) {
            int*   lsrc = (int*)&L[idxs[k]];           // gather address inside LDS
            float* gdst = out + obase + k;
            __builtin_amdgcn_global_store_async_from_lds_b32(
                (gi32_p)(int*)gdst, (li32_p)lsrc, 0, 0);
            if (it == N_ITERS - 1)
                __builtin_amdgcn_global_store_async_from_lds_b32(
                    (gi32_p)(int*)(gdst + odup), (li32_p)lsrc, 0, 0);
        }
        // last iteration: S_ENDPGM performs an implicit wait-idle (all counters)
#else
#pragma unroll
        for (int k = tid; k < N_INFO; k += TPB) {
            const int   idx = idxs[k];
            const float u   = L[idx] + R[idx];
            out[obase + k] = u;
            if (it == N_ITERS - 1)
                out[obase + odup + k] = u;
        }
#endif
    }
}

extern "C" void kernel_launch(void* const* d_in, const int* in_sizes, int n_in,
                              void* d_out, int out_size, void* d_ws, size_t ws_size,
                              hipStream_t stream) {
    const float* rx       = (const float*)d_in[0];
    const int*   info_idx = (const int*)d_in[1];
    float*       out      = (float*)d_out;

    const int batch = in_sizes[0] / N_CODE;   // 2048

    // 22 * 1024 f32 (left+right state) + 512 i32 (info indices) = 92,160 B of LDS
    const size_t smem_bytes = (size_t)(22 * N_CODE) * sizeof(float) + N_INFO * sizeof(int);
    (void)hipFuncSetAttribute((const void*)polar_bp_fg_kernel,
                              hipFuncAttributeMaxDynamicSharedMemorySize, (int)smem_bytes);

    polar_bp_fg_kernel<<<batch, TPB, smem_bytes, stream>>>(rx, info_idx, out, batch);
}